// LayerLoss_12292196401295
// MI455X (gfx1250) — compile-verified
//
#include <hip/hip_runtime.h>
#include <math.h>

// ---------------------------------------------------------------------------
// Problem constants (from the reference)
// ---------------------------------------------------------------------------
#define B_ROWS   512
#define C_CLS    1000
#define T_TENSORS 64
#define P_ELEMS  500000          // divisible by 4 -> clean float4 streaming
#define NW       9

#define BLOCKS_PER_TENSOR 64     // 64*64 = 4096 blocks, 1M threads total
#define BLKT 256                 // 8 wave32 per block

// Workspace (float) layout: [0..4] = ce,hinge,kl,entropy,energy_sq
//                           [8..8+T*4) = per-tensor {l1z, l2z_sq, l1p, l2p_sq}
#define WS_GLOB   0
#define WS_PARAM  8
#define WS_FLOATS (WS_PARAM + T_TENSORS * 4)

typedef __attribute__((ext_vector_type(2))) float v2f;
typedef __attribute__((ext_vector_type(4))) float f32x4;
typedef __attribute__((ext_vector_type(8))) float v8f;

// ---------------------------------------------------------------------------
// Exact f32 wave32 reduction via V_WMMA_F32_16X16X4_F32.
// A (16x4) layout: lane L<16 -> A[L][0],A[L][1]; lane L>=16 -> A[L-16][2],A[L-16][3].
// We place (s, 0) so A[m][0]=s_m, A[m][2]=s_{m+16}, others 0.  B = all-ones
// (layout-independent).  D[m][n] = s_m + s_{m+16}, replicated across n.
// C/D layout: VGPR r holds D[r][lane] (lanes 0-15) / D[r+8][lane-16] (16-31),
// so per-lane sum of the 8 C VGPRs + shfl_xor(16) add = full 32-lane sum.
// ---------------------------------------------------------------------------
__device__ __forceinline__ float wmma_wave_sum(float s) {
  v2f a; a[0] = s;    a[1] = 0.0f;
  v2f b; b[0] = 1.0f; b[1] = 1.0f;
  v8f c = {};
  c = __builtin_amdgcn_wmma_f32_16x16x4_f32(
      /*neg_a=*/false, a, /*neg_b=*/false, b,
      /*c_mod=*/(short)0, c, /*reuse_a=*/false, /*reuse_b=*/false);
  float r = c[0] + c[1] + c[2] + c[3] + c[4] + c[5] + c[6] + c[7];
  r += __shfl_xor(r, 16, 32);
  return r;                      // all lanes hold the wave total
}

// ---------------------------------------------------------------------------
// Kernel 0: zero the accumulator workspace (harness poisons it with 0xAA).
// ---------------------------------------------------------------------------
__global__ void zero_ws_kernel(float* __restrict__ ws) {
  int i = blockIdx.x * blockDim.x + threadIdx.x;
  if (i < WS_FLOATS) ws[i] = 0.0f;
}

// ---------------------------------------------------------------------------
// Kernel 1: per-row softmax losses.  One 256-thread block per row; row cached
// in LDS (4 KB).  Accumulates 5 global scalars with atomics.
// ---------------------------------------------------------------------------
__global__ void logits_losses_kernel(const float* __restrict__ outp,
                                     const int*   __restrict__ tgt,
                                     float*       __restrict__ ws) {
  __shared__ float row[C_CLS];
  __shared__ float red4[8][4];
  __shared__ float red1[8];

  const int r = blockIdx.x;
  const float* o = outp + (size_t)r * C_CLS;
  const int tr = tgt[r];

  for (int c = threadIdx.x; c < C_CLS; c += blockDim.x) row[c] = o[c];
  __syncthreads();
  const float ot = row[tr];

  const int lane = threadIdx.x & 31;
  const int wid  = threadIdx.x >> 5;

  // ---- pass 1: row max ----
  float m = -INFINITY;
  for (int c = threadIdx.x; c < C_CLS; c += blockDim.x) m = fmaxf(m, row[c]);
  for (int off = 16; off; off >>= 1) m = fmaxf(m, __shfl_xor(m, off, 32));
  if (lane == 0) red1[wid] = m;
  __syncthreads();
  m = red1[0];
  for (int w = 1; w < 8; ++w) m = fmaxf(m, red1[w]);
  __syncthreads();

  // ---- pass 2: sum exp / sum o / sum o^2 / hinge ----
  float sumexp = 0.0f, sum_o = 0.0f, sum_sq = 0.0f, hinge = 0.0f;
  for (int c = threadIdx.x; c < C_CLS; c += blockDim.x) {
    float v = row[c];
    sumexp += expf(v - m);
    sum_o  += v;
    sum_sq = fmaf(v, v, sum_sq);
    float mg = 1.0f - ot + v;
    if (c != tr && mg > 0.0f) hinge += mg;
  }
  for (int off = 16; off; off >>= 1) {
    sumexp += __shfl_xor(sumexp, off, 32);
    sum_o  += __shfl_xor(sum_o,  off, 32);
    sum_sq += __shfl_xor(sum_sq, off, 32);
    hinge  += __shfl_xor(hinge,  off, 32);
  }
  if (lane == 0) {
    red4[wid][0] = sumexp; red4[wid][1] = sum_o;
    red4[wid][2] = sum_sq; red4[wid][3] = hinge;
  }
  __syncthreads();
  float t_sumexp = 0.0f, t_sumo = 0.0f, t_sumsq = 0.0f, t_hinge = 0.0f;
  for (int w = 0; w < 8; ++w) {
    t_sumexp += red4[w][0]; t_sumo += red4[w][1];
    t_sumsq  += red4[w][2]; t_hinge += red4[w][3];
  }
  __syncthreads();
  const float lse = m + logf(t_sumexp);        // uniform across block

  // ---- pass 3: entropy  sum probs*logp = sum exp(lp)*lp ----
  float ent = 0.0f;
  for (int c = threadIdx.x; c < C_CLS; c += blockDim.x) {
    float lp = row[c] - lse;
    ent = fmaf(expf(lp), lp, ent);
  }
  for (int off = 16; off; off >>= 1) ent += __shfl_xor(ent, off, 32);
  if (lane == 0) red1[wid] = ent;
  __syncthreads();

  if (threadIdx.x == 0) {
    float t_ent = 0.0f;
    for (int w = 0; w < 8; ++w) t_ent += red1[w];
    const float invB = 1.0f / (float)B_ROWS;
    const float invC = 1.0f / (float)C_CLS;
    atomicAdd(&ws[WS_GLOB + 0], (lse - ot) * invB);                 // CE
    atomicAdd(&ws[WS_GLOB + 1], t_hinge * invB * invC);             // hinge
    float klr = -logf((float)C_CLS) - t_sumo * invC + lse;          // row KL
    atomicAdd(&ws[WS_GLOB + 2], klr * invB);
    atomicAdd(&ws[WS_GLOB + 3], -t_ent * invB);                     // entropy
    atomicAdd(&ws[WS_GLOB + 4], t_sumsq);                           // energy^2
  }
}

// ---------------------------------------------------------------------------
// Kernel 2: streaming param norms.  Dominant cost: 256 MB of read-once HBM
// traffic -> b128 non-temporal loads, grid-stride within each tensor, f32
// per-thread accumulators, then exact WMMA wave reduction + LDS + atomics.
// ---------------------------------------------------------------------------
__global__ void param_norms_kernel(const float* __restrict__ p,
                                   const float* __restrict__ q,
                                   float*       __restrict__ acc /* [T*4] */) {
  const int t   = blockIdx.x / BLOCKS_PER_TENSOR;
  const int blk = blockIdx.x % BLOCKS_PER_TENSOR;
  const size_t base = (size_t)t * P_ELEMS;
  const f32x4* pv = reinterpret_cast<const f32x4*>(p + base);
  const f32x4* qv = reinterpret_cast<const f32x4*>(q + base);

  const int nvec   = P_ELEMS / 4;                    // 125000 float4 per tensor
  const int stride = BLOCKS_PER_TENSOR * BLKT;       // 16384

  float s0 = 0.0f, s1 = 0.0f, s2 = 0.0f, s3 = 0.0f;
  for (int i = blk * BLKT + threadIdx.x; i < nvec; i += stride) {
    f32x4 a = __builtin_nontemporal_load(pv + i);    // read-once: NT hint
    f32x4 b = __builtin_nontemporal_load(qv + i);
    s0 += fabsf(a[0]) + fabsf(a[1]) + fabsf(a[2]) + fabsf(a[3]);
    s1 = fmaf(a[0], a[0], fmaf(a[1], a[1], fmaf(a[2], a[2], fmaf(a[3], a[3], s1))));
    float d0 = a[0] - b[0], d1 = a[1] - b[1], d2 = a[2] - b[2], d3 = a[3] - b[3];
    s2 += fabsf(d0) + fabsf(d1) + fabsf(d2) + fabsf(d3);
    s3 = fmaf(d0, d0, fmaf(d1, d1, fmaf(d2, d2, fmaf(d3, d3, s3))));
  }

  // Exact f32 wave reduction on the matrix pipe (EXEC all-ones: unconditional)
  float r0 = wmma_wave_sum(s0);
  float r1 = wmma_wave_sum(s1);
  float r2 = wmma_wave_sum(s2);
  float r3 = wmma_wave_sum(s3);

  __shared__ float wred[8][4];
  const int lane = threadIdx.x & 31;
  const int wid  = threadIdx.x >> 5;
  if (lane == 0) { wred[wid][0] = r0; wred[wid][1] = r1;
                   wred[wid][2] = r2; wred[wid][3] = r3; }
  __syncthreads();
  if (threadIdx.x == 0) {
    float a0 = 0.0f, a1 = 0.0f, a2 = 0.0f, a3 = 0.0f;
    for (int w = 0; w < 8; ++w) {
      a0 += wred[w][0]; a1 += wred[w][1]; a2 += wred[w][2]; a3 += wred[w][3];
    }
    atomicAdd(&acc[t * 4 + 0], a0);
    atomicAdd(&acc[t * 4 + 1], a1);
    atomicAdd(&acc[t * 4 + 2], a2);
    atomicAdd(&acc[t * 4 + 3], a3);
  }
}

// ---------------------------------------------------------------------------
// Kernel 3: finalize.  64 threads (one per tensor), 2 waves.
// ---------------------------------------------------------------------------
__global__ void finalize_kernel(const float* __restrict__ ws,
                                const float* __restrict__ lw,
                                float*       __restrict__ out) {
  __shared__ float red[2];
  const int t = threadIdx.x;

  float comb[NW];
  comb[0] = ws[WS_GLOB + 0];            // ce
  comb[1] = ws[WS_GLOB + 1];            // hinge
  comb[2] = ws[WS_GLOB + 2];            // kl
  comb[3] = ws[WS_GLOB + 3];            // entropy
  comb[4] = sqrtf(ws[WS_GLOB + 4]);     // energy
  const float* pa = ws + WS_PARAM + t * 4;
  comb[5] = pa[0];                      // l1_zero
  comb[6] = sqrtf(pa[1]);               // l2_zero
  comb[7] = pa[2];                      // l1_pre
  comb[8] = sqrtf(pa[3]);               // l2_pre

  float s = 0.0f;
  #pragma unroll
  for (int j = 0; j < NW; ++j) {
    float w   = lw[t * NW + j];
    float sig = 1.0f / (1.0f + expf(-w));
    s = fmaf(sig, comb[j], s);
  }
  for (int off = 16; off; off >>= 1) s += __shfl_xor(s, off, 32);
  if ((threadIdx.x & 31) == 0) red[threadIdx.x >> 5] = s;
  __syncthreads();
  if (threadIdx.x == 0)
    out[0] = (red[0] + red[1]) * (1.0f / ((float)T_TENSORS * (float)NW));
}

// ---------------------------------------------------------------------------
// Launch: init -> logits -> params -> finalize, all stream-ordered
// (graph-capture safe, no sync APIs, deterministic work per call).
// ---------------------------------------------------------------------------
extern "C" void kernel_launch(void* const* d_in, const int* in_sizes, int n_in,
                              void* d_out, int out_size, void* d_ws, size_t ws_size,
                              hipStream_t stream) {
  const float* outputs = (const float*)d_in[0];   // [512,1000] f32
  const int*   targets = (const int*)  d_in[1];   // [512] i32
  const float* params  = (const float*)d_in[2];   // [64,500000] f32
  const float* pretr   = (const float*)d_in[3];   // [64,500000] f32
  const float* lw      = (const float*)d_in[4];   // [64,9] f32
  float* out = (float*)d_out;
  float* ws  = (float*)d_ws;

  zero_ws_kernel<<<1, 512, 0, stream>>>(ws);
  logits_losses_kernel<<<B_ROWS, BLKT, 0, stream>>>(outputs, targets, ws);
  param_norms_kernel<<<T_TENSORS * BLOCKS_PER_TENSOR, BLKT, 0, stream>>>(
      params, pretr, ws + WS_PARAM);
  finalize_kernel<<<1, 64, 0, stream>>>(ws, lw, out);
}